// FSRS_45818711114424
// MI455X (gfx1250) — compile-verified
//
#include <hip/hip_runtime.h>

#define SEQ_LEN 64
#define BLK 256

typedef __attribute__((ext_vector_type(2))) float v2f;   // native vector for NT b64 stores

// FSRS scan: one thread per batch lane, 64 sequential steps, state in VGPRs.
// Next step's inputs are prefetched via the gfx1250 async global->LDS path
// (double-buffered per-lane staging, per-wave ASYNCcnt wait, no barriers).
// Output stream uses non-temporal stores so the input stays L2-resident
// across graph replays (input 64MB + output 68MB vs 192MB L2).
__global__ __launch_bounds__(BLK) void fsrs_scan_kernel(
    const float* __restrict__ X,   // (SEQ_LEN, batch, 2)
    const float* __restrict__ w,   // (17,)
    float* __restrict__ out,       // (SEQ_LEN, batch, 2) then (batch, 2)
    int batch)
{
    __shared__ float tile[2][2 * BLK];   // double-buffered per-lane staging (4 KB)

    const int tid = threadIdx.x;
    const long long i = (long long)blockIdx.x * BLK + tid;
    if (i >= batch) return;

    // Uniform weight loads -> scalarized by compiler (s_load), broadcast free.
    const float w0 = w[0],  w1 = w[1],  w2 = w[2],  w3 = w[3];
    const float w4 = w[4],  w5 = w[5],  w6 = w[6],  w7 = w[7];
    const float w9 = w[9],  w10 = w[10], w11 = w[11], w12 = w[12];
    const float w13 = w[13], w14 = w[14], w15 = w[15], w16 = w[16];
    const float expw8 = __expf(w[8]);

    const v2f* Xp = (const v2f*)X;
    v2f*       Op = (v2f*)out;
    const long long stride = batch;      // v2f elements per step

    // LDS byte offsets of this lane's two staging slots.
    const unsigned int lds0 = (unsigned int)(unsigned long long)&tile[0][2 * tid];
    const unsigned int lds1 = (unsigned int)(unsigned long long)&tile[1][2 * tid];

    // Kick off async prefetch of step-1 inputs into buffer 0 (ASYNCcnt++).
    {
        unsigned long long g = (unsigned long long)&Xp[stride + i];
        asm volatile("global_load_async_to_lds_b64 %0, %1, off"
                     :: "v"(lds0), "v"(g) : "memory");
    }

    // ---- init state from step 0 (direct load, overlapped with async) ----
    const float rating0 = X[2 * i + 1];
    float s, d;
    {
        int ri = (int)rating0 - 1;
        ri = ri < 0 ? 0 : (ri > 3 ? 3 : ri);
        const float sv = (ri == 0) ? w0 : (ri == 1) ? w1 : (ri == 2) ? w2 : w3;
        const bool in_range = (rating0 >= 1.0f) && (rating0 <= 4.0f);
        s = in_range ? sv : 1.0f;
        d = w4 - w5 * (rating0 - 3.0f);
        d = fminf(fmaxf(d, 1.0f), 10.0f);
        s = fminf(fmaxf(s, 0.1f), 36500.0f);
    }

    {
        v2f st; st.x = s; st.y = d;
        __builtin_nontemporal_store(st, &Op[i]);
    }

    // ---- 63 scan steps, async double-buffered input pipeline ----
    #pragma unroll 1
    for (int k = 1; k < SEQ_LEN; ++k) {
        if (k + 1 < SEQ_LEN) {
            // Prefetch step k+1 into the other buffer, then wait only for
            // the older load (ASYNCcnt <= 1): compute/load overlap.
            const unsigned int nl = (k & 1) ? lds1 : lds0;
            unsigned long long g =
                (unsigned long long)&Xp[(long long)(k + 1) * stride + i];
            asm volatile("global_load_async_to_lds_b64 %0, %1, off"
                         :: "v"(nl), "v"(g) : "memory");
            asm volatile("s_wait_asynccnt 0x1" ::: "memory");
        } else {
            asm volatile("s_wait_asynccnt 0x0" ::: "memory");
        }

        // Direct addrspace(3) indexing -> ds_load_b64 (no flat fallback).
        float t, rating;
        if (k & 1) {
            t      = tile[0][2 * tid];
            rating = tile[0][2 * tid + 1];
        } else {
            t      = tile[1][2 * tid];
            rating = tile[1][2 * tid + 1];
        }

        // r = 1 / (1 + t / (9 s))
        const float r = 1.0f / (1.0f + t / (9.0f * s));

        float nd = d - w6 * (rating - 3.0f);
        nd = w7 * w4 + (1.0f - w7) * nd;
        nd = fminf(fmaxf(nd, 1.0f), 10.0f);

        const float hard = (rating == 2.0f) ? w15 : 1.0f;
        const float easy = (rating == 4.0f) ? w16 : 1.0f;
        const float omr  = 1.0f - r;

        // s > 0.1 and nd >= 1 are guaranteed by clips -> fast pow is safe.
        const float s_succ = s * (1.0f + expw8 * (11.0f - nd)
                                  * __powf(s, -w9)
                                  * (__expf(omr * w10) - 1.0f)
                                  * hard * easy);
        const float s_fail = w11 * __powf(nd, -w12)
                                 * (__powf(s + 1.0f, w13) - 1.0f)
                                 * __expf(omr * w14);

        float ns = (rating > 1.0f) ? s_succ : s_fail;
        ns = fminf(fmaxf(ns, 0.1f), 36500.0f);
        s = ns;
        d = nd;

        v2f st; st.x = s; st.y = d;
        __builtin_nontemporal_store(st, &Op[(long long)k * stride + i]);
    }

    // final_state, concatenated after the (SEQ_LEN, batch, 2) outputs
    v2f st; st.x = s; st.y = d;
    __builtin_nontemporal_store(st, &Op[(long long)SEQ_LEN * stride + i]);
}

extern "C" void kernel_launch(void* const* d_in, const int* in_sizes, int n_in,
                              void* d_out, int out_size, void* d_ws, size_t ws_size,
                              hipStream_t stream) {
    const float* X = (const float*)d_in[0];   // (64, batch, 2) float32
    const float* w = (const float*)d_in[1];   // (17,) float32
    float* out = (float*)d_out;               // (64*batch*2 + batch*2) float32

    const int batch = in_sizes[0] / (2 * SEQ_LEN);   // 131072
    const int blocks = (batch + BLK - 1) / BLK;       // 512
    fsrs_scan_kernel<<<blocks, BLK, 0, stream>>>(X, w, out, batch);
}